// CQCLoss_70815420776847
// MI455X (gfx1250) — compile-verified
//
#include <hip/hip_runtime.h>
#include <hip/hip_bf16.h>

typedef __attribute__((ext_vector_type(2))) float v2f;
typedef __attribute__((ext_vector_type(8))) float v8f;

#define DIM   256
#define NROW  8192
#define HALFN 4096
#define INV_TAU 2.0f   // 1/0.5

// ---------------------------------------------------------------------------
// Kernel 1: row L2-normalize. One wave (32 lanes) per row, 8 floats per lane.
// ---------------------------------------------------------------------------
__global__ __launch_bounds__(256) void cqc_normalize(const float* __restrict__ Xa,
                                                     const float* __restrict__ Za,
                                                     float* __restrict__ xn) {
    const int w = threadIdx.x >> 5;
    const int l = threadIdx.x & 31;
    const int row = blockIdx.x * 8 + w;                 // 1024 blocks * 8 rows
    const float* src = (row < HALFN) ? (Xa + (size_t)row * DIM)
                                     : (Za + (size_t)(row - HALFN) * DIM);
    float4 v0 = *(const float4*)(src + l * 8);
    float4 v1 = *(const float4*)(src + l * 8 + 4);
    float ss = v0.x*v0.x + v0.y*v0.y + v0.z*v0.z + v0.w*v0.w
             + v1.x*v1.x + v1.y*v1.y + v1.z*v1.z + v1.w*v1.w;
    ss += __shfl_xor(ss, 1);
    ss += __shfl_xor(ss, 2);
    ss += __shfl_xor(ss, 4);
    ss += __shfl_xor(ss, 8);
    ss += __shfl_xor(ss, 16);
    const float inv = 1.0f / fmaxf(sqrtf(ss), 1e-8f);
    v0.x *= inv; v0.y *= inv; v0.z *= inv; v0.w *= inv;
    v1.x *= inv; v1.y *= inv; v1.z *= inv; v1.w *= inv;
    float* dst = xn + (size_t)row * DIM + l * 8;
    *(float4*)(dst)     = v0;
    *(float4*)(dst + 4) = v1;
}

// ---------------------------------------------------------------------------
// Kernel 2: fused fp32 WMMA GEMM tile + exp row-sum.
// Block = 256 threads (8 waves). Block tile = 128 rows x 128 cols.
// Wave (wr,wc): wr in 0..1 -> 64 rows, wc in 0..3 -> 32 cols
//   => 4 (m) x 2 (n) accumulators of V_WMMA_F32_16X16X4_F32.
// Fragment layout (ISA 7.12.2, 32-bit A 16x4 / B 4x16): lane l, vgpr v holds
// K = 2*(l>>4) + v, row/col = l&15  => one float2 load per fragment per K-step.
// Register double-buffering: fragments for step k+4 are issued BEFORE the 8
// WMMAs of step k, so L2 load latency overlaps matrix issue instead of
// serializing on s_wait_loadcnt 0 per WMMA.
// ---------------------------------------------------------------------------
__global__ __launch_bounds__(256) void cqc_sim_expsum(const float* __restrict__ xn,
                                                      float* __restrict__ partial) {
    const int tid  = threadIdx.x;
    const int l    = tid & 31;
    const int w    = tid >> 5;      // 0..7
    const int wr   = w >> 2;        // 0..1  (row band of 64)
    const int wc   = w & 3;         // 0..3  (col band of 32)
    const int lmod  = l & 15;
    const int lhalf = l >> 4;

    const int m_base = blockIdx.y * 128 + wr * 64;
    const int n_base = blockIdx.x * 128 + wc * 32;

    const float* aptr[4];
    #pragma unroll
    for (int mb = 0; mb < 4; ++mb)
        aptr[mb] = xn + (size_t)(m_base + mb * 16 + lmod) * DIM + 2 * lhalf;
    const float* bptr[2];
    #pragma unroll
    for (int nb = 0; nb < 2; ++nb)
        bptr[nb] = xn + (size_t)(n_base + nb * 16 + lmod) * DIM + 2 * lhalf;

    v8f acc[4][2];
    #pragma unroll
    for (int mb = 0; mb < 4; ++mb)
        #pragma unroll
        for (int nb = 0; nb < 2; ++nb)
            acc[mb][nb] = (v8f){0.f,0.f,0.f,0.f,0.f,0.f,0.f,0.f};

    // ---- double-buffered fragment registers ----
    v2f abuf[2][4], bbuf[2][2];
    #pragma unroll
    for (int mb = 0; mb < 4; ++mb) abuf[0][mb] = *(const v2f*)(aptr[mb]);
    #pragma unroll
    for (int nb = 0; nb < 2; ++nb) bbuf[0][nb] = *(const v2f*)(bptr[nb]);

    // steady state: prefetch k+4 into the other buffer, then 8 WMMAs on k
    #pragma unroll 2
    for (int k = 0; k < DIM - 4; k += 4) {
        const int cur = (k >> 2) & 1;
        const int nxt = cur ^ 1;
        #pragma unroll
        for (int mb = 0; mb < 4; ++mb)
            abuf[nxt][mb] = *(const v2f*)(aptr[mb] + k + 4);
        #pragma unroll
        for (int nb = 0; nb < 2; ++nb)
            bbuf[nxt][nb] = *(const v2f*)(bptr[nb] + k + 4);
        #pragma unroll
        for (int mb = 0; mb < 4; ++mb)
            #pragma unroll
            for (int nb = 0; nb < 2; ++nb)
                acc[mb][nb] = __builtin_amdgcn_wmma_f32_16x16x4_f32(
                    false, abuf[cur][mb], false, bbuf[cur][nb], (short)0,
                    acc[mb][nb], false, false);
    }
    // epilogue: last K-step, buffer ((DIM-4)>>2)&1 == 1
    #pragma unroll
    for (int mb = 0; mb < 4; ++mb)
        #pragma unroll
        for (int nb = 0; nb < 2; ++nb)
            acc[mb][nb] = __builtin_amdgcn_wmma_f32_16x16x4_f32(
                false, abuf[1][mb], false, bbuf[1][nb], (short)0,
                acc[mb][nb], false, false);

    // exp + per-row reduction. lsum[rowInBlock][slot]; slot = wc*2+nb (8 slots)
    __shared__ float lsum[128][8];
    #pragma unroll
    for (int mb = 0; mb < 4; ++mb) {
        #pragma unroll
        for (int nb = 0; nb < 2; ++nb) {
            #pragma unroll
            for (int g = 0; g < 8; ++g) {
                float v = __expf(acc[mb][nb][g] * INV_TAU);
                // sum over 16 columns: lanes within each half hold one row
                v += __shfl_xor(v, 1);
                v += __shfl_xor(v, 2);
                v += __shfl_xor(v, 4);
                v += __shfl_xor(v, 8);
                if (lmod == 0) {
                    const int rowInBlk = wr * 64 + mb * 16 + g + 8 * lhalf;
                    lsum[rowInBlk][wc * 2 + nb] = v;
                }
            }
        }
    }
    __syncthreads();

    if (tid < 128) {
        float s = 0.f;
        #pragma unroll
        for (int k = 0; k < 8; ++k) s += lsum[tid][k];
        const int row = blockIdx.y * 128 + tid;
        partial[(size_t)row * 64 + blockIdx.x] = s;   // gridDim.x == 64
    }
}

// ---------------------------------------------------------------------------
// Kernel 3: per-row loss. One block (256 thr) per row.
//   total = sum of 64 partials; sii = xn_i . xn_i; pos = xn_i . xn_mate
//   loss  = -pos/tau + log(total - exp(sii/tau))
// ---------------------------------------------------------------------------
__global__ __launch_bounds__(256) void cqc_row_loss(const float* __restrict__ xn,
                                                    const float* __restrict__ partial,
                                                    float* __restrict__ rowloss) {
    const int i = blockIdx.x;
    const int mate = (i < HALFN) ? (i + HALFN) : (i - HALFN);
    const int t = threadIdx.x;

    const float xi = xn[(size_t)i * DIM + t];
    const float xm = xn[(size_t)mate * DIM + t];
    float ss = xi * xi;
    float pp = xi * xm;
    float pt = (t < 64) ? partial[(size_t)i * 64 + t] : 0.f;

    __shared__ float r0[256], r1[256], r2[256];
    r0[t] = ss; r1[t] = pp; r2[t] = pt;
    __syncthreads();
    for (int st = 128; st > 0; st >>= 1) {
        if (t < st) {
            r0[t] += r0[t + st];
            r1[t] += r1[t + st];
            r2[t] += r2[t + st];
        }
        __syncthreads();
    }
    if (t == 0) {
        const float sii   = r0[0];
        const float pos   = r1[0];
        const float total = r2[0];
        const float denom = total - __expf(sii * INV_TAU);
        rowloss[i] = -pos * INV_TAU + logf(denom);
    }
}

// ---------------------------------------------------------------------------
// Kernel 4: final reduction of 8192 row losses -> mean.
// ---------------------------------------------------------------------------
__global__ __launch_bounds__(256) void cqc_final(const float* __restrict__ rowloss,
                                                 float* __restrict__ out) {
    __shared__ float s[256];
    float v = 0.f;
    for (int i = threadIdx.x; i < NROW; i += 256) v += rowloss[i];
    s[threadIdx.x] = v;
    __syncthreads();
    for (int st = 128; st > 0; st >>= 1) {
        if (threadIdx.x < st) s[threadIdx.x] += s[threadIdx.x + st];
        __syncthreads();
    }
    if (threadIdx.x == 0) out[0] = s[0] / (float)NROW;
}

extern "C" void kernel_launch(void* const* d_in, const int* in_sizes, int n_in,
                              void* d_out, int out_size, void* d_ws, size_t ws_size,
                              hipStream_t stream) {
    const float* Xa = (const float*)d_in[0];
    const float* Za = (const float*)d_in[1];
    float* out = (float*)d_out;

    float* xn      = (float*)d_ws;                       // 8192*256 = 2M floats
    float* partial = xn + (size_t)NROW * DIM;            // 8192*64  = 512K floats
    float* rowloss = partial + (size_t)NROW * 64;        // 8192 floats

    cqc_normalize<<<NROW / 8, 256, 0, stream>>>(Xa, Za, xn);

    dim3 grid2(NROW / 128, NROW / 128);                  // 64 x 64
    cqc_sim_expsum<<<grid2, 256, 0, stream>>>(xn, partial);

    cqc_row_loss<<<NROW, 256, 0, stream>>>(xn, partial, rowloss);

    cqc_final<<<1, 256, 0, stream>>>(rowloss, out);
}